// my_GCN_23605140259107
// MI455X (gfx1250) — compile-verified
//
#include <hip/hip_runtime.h>
#include <hip/hip_bf16.h>

#define N_NODES 100000
#define N_EDGES 1600000
#define FEATS   128

typedef __attribute__((ext_vector_type(2))) float v2f;
typedef __attribute__((ext_vector_type(8))) float v8f;

// ---------------- degree + norm ----------------
__global__ void degree_kernel(const int* __restrict__ src, const int* __restrict__ dst,
                              int* __restrict__ outdeg, int* __restrict__ indeg, int n) {
    int i = blockIdx.x * blockDim.x + threadIdx.x;
    if (i < n) {
        atomicAdd(&outdeg[src[i]], 1);
        atomicAdd(&indeg[dst[i]], 1);
    }
}

__global__ void norm_kernel(const int* __restrict__ outdeg, const int* __restrict__ indeg,
                            float* __restrict__ out_norm, float* __restrict__ in_norm, int n) {
    int i = blockIdx.x * blockDim.x + threadIdx.x;
    if (i < n) {
        int od = outdeg[i]; if (od < 1) od = 1;
        int id = indeg[i];  if (id < 1) id = 1;
        out_norm[i] = rsqrtf((float)od);
        in_norm[i]  = rsqrtf((float)id);
    }
}

// ---------------- edge scatter: agg[dst] += x[src] * out_norm[src] ----------------
// one wave32 per edge; lane handles 4 consecutive floats (32*4 = 128)
__global__ void scatter_kernel(const float* __restrict__ x,
                               const int* __restrict__ src,
                               const int* __restrict__ dst,
                               const float* __restrict__ out_norm,
                               float* __restrict__ agg, int n_edges) {
    int warp = (blockIdx.x * blockDim.x + threadIdx.x) >> 5;
    int lane = threadIdx.x & 31;
    if (warp >= n_edges) return;
    int s = src[warp];
    int d = dst[warp];
    float sc = out_norm[s];
    const float4* xs = (const float4*)(x + (size_t)s * FEATS);
    float4 v = xs[lane];
    float* ad = agg + (size_t)d * FEATS + lane * 4;
    atomicAdd(ad + 0, v.x * sc);
    atomicAdd(ad + 1, v.y * sc);
    atomicAdd(ad + 2, v.z * sc);
    atomicAdd(ad + 3, v.w * sc);
}

// ---------------- WMMA f32 GEMM: C[M,N] = rowscale(A[M,K]) @ B (+bias, relu) ----------------
// One wave per 16x16 C tile, K-loop in steps of 4 using V_WMMA_F32_16X16X4_F32.
// A 16x4 layout:  lane<16 -> row=lane, k={0,1}; lane>=16 -> row=lane-16, k={2,3}
// B 4x16 layout:  lane<16 -> col=lane, k={0,1}; lane>=16 -> col=lane-16, k={2,3}
// C 16x16 layout: vgpr r -> M = r + 8*(lane>>4), N = lane&15
template<bool BT, bool RELU, bool HAS_SCALE, bool HAS_BIAS>
__global__ void gemm_wmma_kernel(const float* __restrict__ A,
                                 const float* __restrict__ B,
                                 const float* __restrict__ rowscale,
                                 const float* __restrict__ bias,
                                 float* __restrict__ C,
                                 int M, int N, int K, int tiles_n) {
    int wave = (int)((blockIdx.x * blockDim.x + threadIdx.x) >> 5);
    int lane = threadIdx.x & 31;
    int total_tiles = (M >> 4) * tiles_n;
    if (wave >= total_tiles) return;   // wave-uniform: EXEC stays all-ones for WMMA

    int tile_m = wave / tiles_n;
    int tile_n = wave - tile_m * tiles_n;

    int half = lane >> 4;      // selects k-pair {0,1} vs {2,3}
    int l    = lane & 15;
    int arow = tile_m * 16 + l;
    int bcol = tile_n * 16 + l;

    float scale = HAS_SCALE ? rowscale[arow] : 1.0f;

    const float* Arow = A + (size_t)arow * K;
    v8f c = {};

    for (int k = 0; k < K; k += 4) {
        int kb = k + half * 2;                       // even -> 8B aligned
        v2f a = *(const v2f*)(Arow + kb);
        if (HAS_SCALE) { a.x *= scale; a.y *= scale; }
        v2f b;
        if (BT) {
            b = *(const v2f*)(B + (size_t)bcol * K + kb);  // B[n][k], B[n][k+1]
        } else {
            b.x = B[(size_t)kb * N + bcol];                // B[k][n]
            b.y = B[(size_t)(kb + 1) * N + bcol];
        }
        c = __builtin_amdgcn_wmma_f32_16x16x4_f32(false, a, false, b,
                                                  (short)0, c, false, false);
    }

    float bv = HAS_BIAS ? bias[bcol] : 0.0f;
    size_t base = (size_t)(tile_m * 16 + half * 8) * N + bcol;
    #pragma unroll
    for (int r = 0; r < 8; ++r) {
        float v = c[r] + bv;
        if (RELU) v = fmaxf(v, 0.0f);
        C[base + (size_t)r * N] = v;
    }
}

// ---------------- h = relu(agg * in_norm[row] + b2[col]) ----------------
__global__ void finalize_kernel(const float* __restrict__ agg,
                                const float* __restrict__ in_norm,
                                const float* __restrict__ b2,
                                float* __restrict__ h, int total) {
    int i = blockIdx.x * blockDim.x + threadIdx.x;
    if (i < total) {
        int row = i >> 7;          // /128
        int col = i & 127;
        h[i] = fmaxf(agg[i] * in_norm[row] + b2[col], 0.0f);
    }
}

extern "C" void kernel_launch(void* const* d_in, const int* in_sizes, int n_in,
                              void* d_out, int out_size, void* d_ws, size_t ws_size,
                              hipStream_t stream) {
    const float* in_feat = (const float*)d_in[0];
    const int*   src     = (const int*)d_in[1];
    const int*   dst     = (const int*)d_in[2];
    const float* W1      = (const float*)d_in[3];   // [128,256]
    const float* b1      = (const float*)d_in[4];   // [256]
    const float* W2      = (const float*)d_in[5];   // [256,128]
    const float* b2      = (const float*)d_in[6];   // [128]
    const float* fc1_w   = (const float*)d_in[7];   // [128,128]
    const float* fc2_w   = (const float*)d_in[8];   // [128,128]

    // d_out segments: [h | feat1 | feat2], each N_NODES*FEATS floats.
    float* seg0 = (float*)d_out;                         // final h
    float* seg1 = seg0 + (size_t)N_NODES * FEATS;        // staging agg1/agg2, final feat1
    float* seg2 = seg1 + (size_t)N_NODES * FEATS;        // staging xw2,       final feat2

    // workspace: degrees, norms, hidden h1 [100k,256]
    int*   outdeg   = (int*)d_ws;
    int*   indeg    = outdeg + N_NODES;
    float* out_norm = (float*)(indeg + N_NODES);
    float* in_norm  = out_norm + N_NODES;
    float* h1       = in_norm + N_NODES;                 // N_NODES * 256 floats

    const size_t row_bytes = (size_t)N_NODES * FEATS * sizeof(float);

    // 1) degrees + norms
    hipMemsetAsync(outdeg, 0, 2 * N_NODES * sizeof(int), stream);
    degree_kernel<<<(N_EDGES + 255) / 256, 256, 0, stream>>>(src, dst, outdeg, indeg, N_EDGES);
    norm_kernel<<<(N_NODES + 255) / 256, 256, 0, stream>>>(outdeg, indeg, out_norm, in_norm, N_NODES);

    // 2) agg1 = scatter(in_feat * out_norm[src])  -> seg1
    hipMemsetAsync(seg1, 0, row_bytes, stream);
    scatter_kernel<<<(N_EDGES * 32 + 255) / 256, 256, 0, stream>>>(
        in_feat, src, dst, out_norm, seg1, N_EDGES);

    // 3) h1 = relu((agg1 * in_norm) @ W1 + b1)   [100k,256]
    {
        int tiles = (N_NODES / 16) * (256 / 16);     // 100000 tiles, 8 waves/block
        gemm_wmma_kernel<false, true, true, true><<<tiles / 8, 256, 0, stream>>>(
            seg1, W1, in_norm, b1, h1, N_NODES, 256, 128, 256 / 16);
    }

    // 4) xw2 = h1 @ W2                            [100k,128] -> seg2
    {
        int tiles = (N_NODES / 16) * (128 / 16);     // 50000 tiles
        gemm_wmma_kernel<false, false, false, false><<<(tiles + 7) / 8, 256, 0, stream>>>(
            h1, W2, nullptr, nullptr, seg2, N_NODES, 128, 256, 128 / 16);
    }

    // 5) agg2 = scatter(xw2 * out_norm[src])     -> seg1 (h1 dead now)
    hipMemsetAsync(seg1, 0, row_bytes, stream);
    scatter_kernel<<<(N_EDGES * 32 + 255) / 256, 256, 0, stream>>>(
        seg2, src, dst, out_norm, seg1, N_EDGES);

    // 6) h = relu(agg2 * in_norm + b2)           -> seg0
    {
        int total = N_NODES * FEATS;
        finalize_kernel<<<(total + 255) / 256, 256, 0, stream>>>(seg1, in_norm, b2, seg0, total);
    }

    // 7) feat1 = h @ fc1_w.T  -> seg1 ; feat2 = h @ fc2_w.T -> seg2
    {
        int tiles = (N_NODES / 16) * (128 / 16);
        gemm_wmma_kernel<true, false, false, false><<<(tiles + 7) / 8, 256, 0, stream>>>(
            seg0, fc1_w, nullptr, nullptr, seg1, N_NODES, 128, 128, 128 / 16);
        gemm_wmma_kernel<true, false, false, false><<<(tiles + 7) / 8, 256, 0, stream>>>(
            seg0, fc2_w, nullptr, nullptr, seg2, N_NODES, 128, 128, 128 / 16);
    }
}